// OWNNorm_25469156065406
// MI455X (gfx1250) — compile-verified
//
#include <hip/hip_runtime.h>
#include <hip/hip_bf16.h>

// ---------------------------------------------------------------------------
// OWN-Norm (grouped ZCA whitening) for MI455X / gfx1250
//   weight [8192,8192] fp32, 64 groups of 128 rows.
//   Pipeline: cov_partial (K-split Z*Z^T + row sums) -> cov_reduce (mean
//   correction) -> newton (LDS-resident Newton-Schulz inv-sqrt) -> whiten.
//   All matmuls use V_WMMA_F32_16X16X4_F32 (fp32 in/out, wave32).
//   Memory-bound problem (~0.8 GB min traffic @ 23.3 TB/s ~ 34 us); fp32
//   WMMA easily covers the 39 GFLOP in that window.
// ---------------------------------------------------------------------------

typedef float v2f __attribute__((ext_vector_type(2)));
typedef float v8f __attribute__((ext_vector_type(8)));

#define GS     128      // rows per group
#define DCOLS  8192     // columns
#define NGRP   64
#define NSEG   8        // K-split for cov pass
#define SEGW   (DCOLS / NSEG)   // 1024 cols per segment
#define CHW    64       // chunk width staged in LDS (cov pass)
#define CHP    68       // padded LDS stride (68 % 64 == 4 -> conflict-free cols)
#define LDP    132      // padded LDS stride for 128-wide tiles (132 % 64 == 4)

// workspace layout (floats)
#define WS_MEANS 0
#define WS_SUM   (NGRP * GS)                         // 8192
#define WS_COVP  (WS_SUM + NGRP * NSEG * GS)         // + 65536
// covp: NGRP * NSEG * GS*GS floats; reduced cov lives in seg0, inv-sqrt in seg1

__device__ inline v8f wmma_f32(v2f a, v2f b, v8f c) {
    return __builtin_amdgcn_wmma_f32_16x16x4_f32(false, a, false, b,
                                                 (short)0, c, false, false);
}

__device__ inline void zero_acc(v8f* acc) {
#pragma unroll
    for (int n = 0; n < 8; ++n)
#pragma unroll
        for (int i = 0; i < 8; ++i) acc[n][i] = 0.0f;
}

// Batched fragment step: issue A + all 8 B LDS loads first, then 8 WMMAs, so
// the scheduler can overlap DS latency with math (partial dscnt waits).
//  A rows from `amat` (row-major, stride astr), B rows from `bmat`:
//    bTransposed=true  : B[kk][nn] = bmat[16n+nn][k+kk]   (Z * Z^T pattern)
//    bTransposed=false : B[kk][nn] = bmat[k+kk][16n+nn]   (normal matmul)
template <bool BT>
__device__ inline void wmma_step(const float* __restrict__ amat, int astr,
                                 const float* __restrict__ bmat, int bstr,
                                 int wv, int m16, int kh, int k, v8f* acc) {
    v2f a;
    a.x = amat[(16 * wv + m16) * astr + k + 2 * kh];
    a.y = amat[(16 * wv + m16) * astr + k + 2 * kh + 1];
    v2f bf[8];
#pragma unroll
    for (int n = 0; n < 8; ++n) {
        if (BT) {
            bf[n].x = bmat[(16 * n + m16) * bstr + k + 2 * kh];
            bf[n].y = bmat[(16 * n + m16) * bstr + k + 2 * kh + 1];
        } else {
            bf[n].x = bmat[(k + 2 * kh) * bstr + 16 * n + m16];
            bf[n].y = bmat[(k + 2 * kh + 1) * bstr + 16 * n + m16];
        }
    }
#pragma unroll
    for (int n = 0; n < 8; ++n) acc[n] = wmma_f32(a, bf[n], acc[n]);
}

// ---------------------------------------------------------------------------
// Kernel 1: per-(group, K-segment) partial Z*Z^T and partial row sums.
// grid (NSEG, NGRP), 256 threads (8 waves, wave w owns output tile-row w).
// ---------------------------------------------------------------------------
__global__ __launch_bounds__(256) void cov_partial_kernel(
        const float* __restrict__ W, float* __restrict__ ws) {
    __shared__ float chunk[GS * CHP];
    __shared__ float sred[256];

    const int seg = blockIdx.x, g = blockIdx.y;
    const int tid = threadIdx.x;
    const int wv = tid >> 5, lane = tid & 31;
    const int m16 = lane & 15, kh = lane >> 4;
    const int grow0 = g * GS;

    v8f acc[8];
    zero_acc(acc);
    float rs = 0.0f;

    for (int ch = 0; ch < SEGW / CHW; ++ch) {
        const int colbase = seg * SEGW + ch * CHW;
        // cooperative load 128x64 fp32 (float4, coalesced)
#pragma unroll
        for (int it = 0; it < 8; ++it) {
            int l = it * 256 + tid;        // 0..2047 float4 slots
            int row = l >> 4;              // 16 float4 per row
            int c4 = l & 15;
            const float4 v = *reinterpret_cast<const float4*>(
                W + (size_t)(grow0 + row) * DCOLS + colbase + c4 * 4);
            float* p = &chunk[row * CHP + c4 * 4];
            p[0] = v.x; p[1] = v.y; p[2] = v.z; p[3] = v.w;
        }
        __syncthreads();

        // partial row sums (for mean correction): thread t sums row t&127,
        // half the columns each.
        {
            int row = tid & 127;
            int cb = (tid >> 7) * 32;
            float s = 0.0f;
#pragma unroll
            for (int j = 0; j < 32; ++j) s += chunk[row * CHP + cb + j];
            rs += s;
        }

        // WMMA: tile (wv, n) of Z*Z^T; A and B both come from `chunk`
        for (int k = 0; k < CHW; k += 4)
            wmma_step<true>(chunk, CHP, chunk, CHP, wv, m16, kh, k, acc);
        __syncthreads();
    }

    float* covp = ws + WS_COVP + (size_t)(g * NSEG + seg) * (GS * GS);
#pragma unroll
    for (int n = 0; n < 8; ++n)
#pragma unroll
        for (int r = 0; r < 8; ++r) {
            int row = 16 * wv + r + 8 * kh;
            int col = 16 * n + m16;
            covp[row * GS + col] = acc[n][r];
        }

    sred[tid] = rs;
    __syncthreads();
    if (tid < 128)
        ws[WS_SUM + (size_t)(g * NSEG + seg) * GS + tid] =
            sred[tid] + sred[tid + 128];
}

// ---------------------------------------------------------------------------
// Kernel 2: reduce K-segment partials; cov = sum(partials) - D*m*m^T.
// Reduced cov is written into seg0's slot. Means saved for the whiten pass.
// ---------------------------------------------------------------------------
__global__ __launch_bounds__(256) void cov_reduce_kernel(float* __restrict__ ws) {
    __shared__ float mean_s[GS];
    const int g = blockIdx.x, tid = threadIdx.x;

    if (tid < GS) {
        float s = 0.0f;
        for (int seg = 0; seg < NSEG; ++seg)
            s += ws[WS_SUM + (size_t)(g * NSEG + seg) * GS + tid];
        float m = s / (float)DCOLS;
        mean_s[tid] = m;
        ws[WS_MEANS + g * GS + tid] = m;
    }
    __syncthreads();

    float* base = ws + WS_COVP + (size_t)g * NSEG * (GS * GS);
    for (int e = tid; e < GS * GS; e += 256) {
        float s = 0.0f;
        for (int seg = 0; seg < NSEG; ++seg) s += base[seg * (GS * GS) + e];
        int i = e >> 7, j = e & 127;
        s -= (float)DCOLS * mean_s[i] * mean_s[j];
        base[e] = s;      // element-wise read-then-write of seg0: safe
    }
}

// ---------------------------------------------------------------------------
// Kernel 3: Newton-Schulz inverse sqrt, fully LDS-resident (Y,Z,T = 198 KB).
// grid NGRP, 256 threads. Result (Z / sqrt(||A||_F)) written into seg1 slot.
// ---------------------------------------------------------------------------
__global__ __launch_bounds__(256) void newton_kernel(float* __restrict__ ws) {
    extern __shared__ float smem[];
    const int SZ = GS * LDP;
    float* Y = smem;
    float* Z = smem + SZ;
    float* T = smem + 2 * SZ;

    const int g = blockIdx.x, tid = threadIdx.x;
    const int wv = tid >> 5, lane = tid & 31;
    const int m16 = lane & 15, kh = lane >> 4;

    float* covg = ws + WS_COVP + (size_t)g * NSEG * (GS * GS);  // seg0 = cov

    // load cov into Y, accumulate Frobenius norm^2 (T used as scratch)
    float ss = 0.0f;
    for (int e = tid; e < GS * GS; e += 256) {
        float v = covg[e];
        Y[(e >> 7) * LDP + (e & 127)] = v;
        ss += v * v;
    }
    T[tid] = ss;
    __syncthreads();
    for (int s = 128; s > 0; s >>= 1) {
        if (tid < s) T[tid] += T[tid + s];
        __syncthreads();
    }
    const float norm = sqrtf(T[0]);
    __syncthreads();

    const float rn = 1.0f / norm;
    for (int e = tid; e < GS * GS; e += 256) {
        int i = e >> 7, j = e & 127;
        Y[i * LDP + j] *= rn;
        Z[i * LDP + j] = (i == j) ? 1.0f : 0.0f;
    }
    __syncthreads();

    v8f acc[8];
    for (int iter = 0; iter < 6; ++iter) {
        // T = 1.5 I - 0.5 * (Z @ Y)
        zero_acc(acc);
        for (int k = 0; k < GS; k += 4)
            wmma_step<false>(Z, LDP, Y, LDP, wv, m16, kh, k, acc);
#pragma unroll
        for (int n = 0; n < 8; ++n)
#pragma unroll
            for (int r = 0; r < 8; ++r) {
                int row = 16 * wv + r + 8 * kh;
                int col = 16 * n + m16;
                T[row * LDP + col] =
                    ((row == col) ? 1.5f : 0.0f) - 0.5f * acc[n][r];
            }
        __syncthreads();

        // Y(rows of wave wv) = Y @ T   (only wave wv reads these Y rows ->
        // safe to update in place once its own accumulation is complete)
        zero_acc(acc);
        for (int k = 0; k < GS; k += 4)
            wmma_step<false>(Y, LDP, T, LDP, wv, m16, kh, k, acc);
#pragma unroll
        for (int n = 0; n < 8; ++n)
#pragma unroll
            for (int r = 0; r < 8; ++r)
                Y[(16 * wv + r + 8 * kh) * LDP + 16 * n + m16] = acc[n][r];

        // Z = T @ Z : every wave reads ALL of Z, so accumulate into regs,
        // barrier, then write back.
        zero_acc(acc);
        for (int k = 0; k < GS; k += 4)
            wmma_step<false>(T, LDP, Z, LDP, wv, m16, kh, k, acc);
        __syncthreads();   // all Z reads complete
#pragma unroll
        for (int n = 0; n < 8; ++n)
#pragma unroll
            for (int r = 0; r < 8; ++r)
                Z[(16 * wv + r + 8 * kh) * LDP + 16 * n + m16] = acc[n][r];
        __syncthreads();
    }

    float* invs = covg + (GS * GS);   // seg1 slot holds cov^{-1/2}
    const float sc = rsqrtf(norm);
    for (int e = tid; e < GS * GS; e += 256)
        invs[e] = Z[(e >> 7) * LDP + (e & 127)] * sc;
}

// ---------------------------------------------------------------------------
// Kernel 4: W_norm = cov^{-1/2} @ (Z - mean). grid (64 col-chunks, 64 groups).
// Traversal is REVERSED vs the cov pass so the tail of W (most recently
// cached in the 192 MB L2 by pass 1) is re-read first; output uses
// non-temporal stores to avoid evicting W from L2.
// ---------------------------------------------------------------------------
__global__ __launch_bounds__(256) void whiten_kernel(
        const float* __restrict__ W, const float* __restrict__ ws,
        float* __restrict__ out) {
    extern __shared__ float smem[];
    float* sS = smem;              // inv-sqrt, 128 x (stride 132)
    float* ch = smem + GS * LDP;   // centered Z chunk, 128 x (stride 132)

    const int g = (NGRP - 1) - blockIdx.y;                // reversed
    const int c0 = ((DCOLS / 128 - 1) - blockIdx.x) * 128; // reversed
    const int tid = threadIdx.x;
    const int wv = tid >> 5, lane = tid & 31;
    const int m16 = lane & 15, kh = lane >> 4;

    const float* invs =
        ws + WS_COVP + (size_t)g * NSEG * (GS * GS) + (GS * GS);  // seg1
    const float* means = ws + WS_MEANS + g * GS;

    for (int e = tid; e < GS * GS; e += 256)
        sS[(e >> 7) * LDP + (e & 127)] = invs[e];

#pragma unroll
    for (int t = 0; t < 16; ++t) {
        int l = t * 256 + tid;     // 0..4095 float4 slots
        int row = l >> 5;          // 32 float4 per row
        int c4 = l & 31;
        const float4 v = *reinterpret_cast<const float4*>(
            W + (size_t)(g * GS + row) * DCOLS + c0 + c4 * 4);
        const float m = means[row];
        float* p = &ch[row * LDP + c4 * 4];
        p[0] = v.x - m; p[1] = v.y - m; p[2] = v.z - m; p[3] = v.w - m;
    }
    __syncthreads();

    v8f acc[8];
    zero_acc(acc);
    for (int k = 0; k < GS; k += 4)
        wmma_step<false>(sS, LDP, ch, LDP, wv, m16, kh, k, acc);

#pragma unroll
    for (int n = 0; n < 8; ++n)
#pragma unroll
        for (int r = 0; r < 8; ++r)
            __builtin_nontemporal_store(
                acc[n][r],
                &out[(size_t)(g * GS + 16 * wv + r + 8 * kh) * DCOLS +
                     c0 + 16 * n + m16]);
}

// ---------------------------------------------------------------------------
extern "C" void kernel_launch(void* const* d_in, const int* in_sizes, int n_in,
                              void* d_out, int out_size, void* d_ws,
                              size_t ws_size, hipStream_t stream) {
    (void)in_sizes; (void)n_in; (void)out_size; (void)ws_size;
    const float* W = (const float*)d_in[0];
    float* out = (float*)d_out;
    float* ws = (float*)d_ws;

    // Pass 1: K-split partial covariances + row sums (512 blocks in flight)
    cov_partial_kernel<<<dim3(NSEG, NGRP), 256, 0, stream>>>(W, ws);
    // Pass 2: reduce partials, mean correction
    cov_reduce_kernel<<<NGRP, 256, 0, stream>>>(ws);
    // Pass 3: Newton-Schulz inverse sqrt in LDS (3 x 128x132 fp32 = 198 KB)
    const int newton_lds = 3 * GS * LDP * (int)sizeof(float);
    hipFuncSetAttribute((const void*)newton_kernel,
                        hipFuncAttributeMaxDynamicSharedMemorySize, newton_lds);
    newton_kernel<<<NGRP, 256, newton_lds, stream>>>(ws);
    // Pass 4: whiten (4096 blocks, reversed traversal for L2 reuse of W)
    const int whiten_lds = 2 * GS * LDP * (int)sizeof(float);
    hipFuncSetAttribute((const void*)whiten_kernel,
                        hipFuncAttributeMaxDynamicSharedMemorySize, whiten_lds);
    whiten_kernel<<<dim3(DCOLS / 128, NGRP), 256, whiten_lds, stream>>>(W, ws, out);
}